// Network_90709709291641
// MI455X (gfx1250) — compile-verified
//
#include <hip/hip_runtime.h>
#include <hip/hip_bf16.h>
#include <stdint.h>

#define NLEV 5
#define TOPK 1000u
#define NCLS 16
#define MAXD 4.135166556742356f   // log(1000/16)

struct Params {
    const float* anch[NLEV];
    const float* cls[NLEV];
    const float* reg[NLEV];
    unsigned n[NLEV];
    unsigned monoOff[NLEV];
    unsigned* mono;                 // [sum n]
    unsigned* hist;                 // [NLEV][4][256]
    unsigned* state;                // [NLEV][4] : prefix, k_rem, cntA, cntE
    unsigned long long* selA;       // [NLEV][1024]
    unsigned* selE;                 // [NLEV][1024]
    unsigned* order;                // [NLEV][TOPK]
};

__device__ __forceinline__ unsigned mono32(float f) {
    unsigned u = __float_as_uint(f);
    return (u & 0x80000000u) ? ~u : (u | 0x80000000u);
}

__global__ void k_init(Params P) {
    unsigned t = threadIdx.x;
    for (unsigned i = t; i < NLEV * 4u * 256u; i += 256u) P.hist[i] = 0u;
    if (t < NLEV) {
        P.state[t * 4 + 0] = 0u;      // prefix
        P.state[t * 4 + 1] = TOPK;    // remaining k
        P.state[t * 4 + 2] = 0u;      // cntA
        P.state[t * 4 + 3] = 0u;      // cntE
    }
}

// Per-anchor max over 16 classes -> monotone u32 key.
// gfx1250: stream cls rows into LDS with global_load_async_to_lds_b128 (ASYNCcnt).
__global__ void k_ruler(Params P) {
    unsigned l = blockIdx.y;
    unsigned n = P.n[l];
    unsigned base = blockIdx.x * 256u;
    if (base >= n) return;
    const float* cls = P.cls[l];
    unsigned* mono = P.mono + P.monoOff[l];
    unsigned t = threadIdx.x;

    __shared__ float4 cbuf[1024];    // 256 rows x 16 floats = 16 KB
    if (base + 256u <= n) {
#if defined(__gfx1250__)
        // Async DMA of the 16 KB tile into LDS: 4 x b128 per lane, 1024 float4 total.
        unsigned long long gbase =
            (unsigned long long)(const void*)cls + (unsigned long long)base * 64ull;
        #pragma unroll
        for (int q = 0; q < 4; ++q) {
            unsigned idx = (unsigned)q * 256u + t;
            unsigned laddr = (unsigned)(unsigned long long)(void*)&cbuf[idx]; // LDS byte offset
            unsigned long long gaddr = gbase + (unsigned long long)idx * 16ull;
            asm volatile("global_load_async_to_lds_b128 %0, %1, off"
                         :: "v"(laddr), "v"(gaddr) : "memory");
        }
        // Prefetch next tile while DMA is in flight (global_prefetch_b8).
        if (base + 512u <= n) __builtin_prefetch(cls + (size_t)(base + 256u + t) * 16u, 0, 1);
        asm volatile("s_wait_asynccnt 0x0" ::: "memory");
        __syncthreads();
#else
        const float4* c4 = (const float4*)cls;
        unsigned b4 = base * 4u;
        #pragma unroll
        for (int q = 0; q < 4; ++q) cbuf[q * 256 + t] = c4[b4 + (unsigned)q * 256u + t];
        __syncthreads();
#endif
        const float* row = (const float*)&cbuf[t * 4];
        float m = -3.402823466e+38f;
        #pragma unroll
        for (int c = 0; c < NCLS; ++c) m = fmaxf(m, row[c]);
        mono[base + t] = mono32(m);
    } else {
        unsigned i = base + t;
        if (i < n) {
            float m = -3.402823466e+38f;
            #pragma unroll
            for (int c = 0; c < NCLS; ++c) m = fmaxf(m, cls[(size_t)i * 16u + c]);
            mono[i] = mono32(m);
        }
    }
}

// Radix-select histogram pass over surviving prefix.
__global__ void k_hist(Params P, int pass) {
    unsigned l = blockIdx.y;
    unsigned n = P.n[l];
    unsigned base = blockIdx.x * 256u;
    if (base >= n) return;
    const unsigned* mono = P.mono + P.monoOff[l];
    unsigned prefix = P.state[l * 4 + 0];
    unsigned maskHigh = pass ? (0xFFFFFFFFu << (32 - 8 * pass)) : 0u;
    int shift = 24 - 8 * pass;

    __shared__ unsigned lh[256];
    unsigned t = threadIdx.x;
    lh[t] = 0u;
    __syncthreads();
    unsigned i = base + t;
    if (i < n) {
        unsigned m = mono[i];
        if ((m & maskHigh) == prefix) atomicAdd(&lh[(m >> shift) & 255u], 1u);
    }
    __syncthreads();
    unsigned c = lh[t];
    if (c) atomicAdd(&P.hist[(l * 4u + (unsigned)pass) * 256u + t], c);
}

// Pick the digit containing the k-th largest element; update prefix/k.
__global__ void k_pick(Params P, int pass) {
    unsigned l = blockIdx.x;
    unsigned* st = P.state + l * 4;
    unsigned k = st[1];
    const unsigned* h = P.hist + (l * 4u + (unsigned)pass) * 256u;
    unsigned acc = 0u;
    int sel = 0;
    for (int d = 255; d >= 0; --d) {
        unsigned c = h[d];
        if (acc + c >= k) { sel = d; break; }
        acc += c;
    }
    st[0] |= ((unsigned)sel) << (24 - 8 * pass);
    st[1] = k - acc;   // ties needed at exact threshold after final pass
}

// Compact keys above threshold + tie indices; block-aggregated global atomics.
__global__ void k_compact(Params P) {
    unsigned l = blockIdx.y;
    unsigned n = P.n[l];
    unsigned base = blockIdx.x * 256u;
    if (base >= n) return;
    unsigned i = base + threadIdx.x;
    unsigned T = P.state[l * 4 + 0];
    unsigned m = (i < n) ? P.mono[P.monoOff[l] + i] : 0u;
    bool isA = (i < n) && (m > T);
    bool isE = (i < n) && (m == T);

    __shared__ unsigned cntLA, cntLE, baseLA, baseLE;
    if (threadIdx.x == 0) { cntLA = 0u; cntLE = 0u; }
    __syncthreads();
    unsigned pA = isA ? atomicAdd(&cntLA, 1u) : 0u;
    unsigned pE = isE ? atomicAdd(&cntLE, 1u) : 0u;
    __syncthreads();
    if (threadIdx.x == 0) {
        baseLA = cntLA ? atomicAdd(&P.state[l * 4 + 2], cntLA) : 0u;
        baseLE = cntLE ? atomicAdd(&P.state[l * 4 + 3], cntLE) : 0u;
    }
    __syncthreads();
    if (isA) {
        unsigned pos = baseLA + pA;
        if (pos < 1024u)
            P.selA[l * 1024u + pos] = ((unsigned long long)m << 32) | (unsigned)(~i);
    }
    if (isE) {
        unsigned pos = baseLE + pE;
        if (pos < 1024u) P.selE[l * 1024u + pos] = i;
    }
}

// Per-level bitonic sort (descending) of <=2048 candidate keys; emit ordered indices.
__global__ void k_sort(Params P) {
    unsigned l = blockIdx.x;
    __shared__ unsigned long long keys[2048];
    const unsigned* st = P.state + l * 4;
    unsigned T = st[0];
    unsigned nA = st[2]; if (nA > 1024u) nA = 1024u;
    unsigned nE = st[3]; if (nE > 1024u) nE = 1024u;
    unsigned t = threadIdx.x;

    for (unsigned j = t; j < 2048u; j += 1024u) {
        unsigned long long key = 0ull;
        if (j < nA) key = P.selA[l * 1024u + j];
        else if (j - nA < nE)
            key = ((unsigned long long)T << 32) | (unsigned)(~P.selE[l * 1024u + (j - nA)]);
        keys[j] = key;
    }
    __syncthreads();

    for (unsigned size = 2u; size <= 2048u; size <<= 1) {
        for (unsigned stride = size >> 1; stride > 0u; stride >>= 1) {
            #pragma unroll
            for (unsigned q = 0; q < 2; ++q) {
                unsigned i = t + q * 1024u;
                unsigned ixj = i ^ stride;
                if (ixj > i) {
                    unsigned long long a = keys[i], b = keys[ixj];
                    bool desc = ((i & size) == 0u);
                    if (desc ? (a < b) : (a > b)) { keys[i] = b; keys[ixj] = a; }
                }
            }
            __syncthreads();
        }
    }
    if (t < TOPK) P.order[l * TOPK + t] = ~(unsigned)keys[t];
}

// Decode boxes, sigmoid scores, tags. One thread per (kept_anchor, class) row.
__global__ void k_out(Params P, float* __restrict__ out, unsigned rows) {
    unsigned tid = blockIdx.x * 256u + threadIdx.x;
    if (tid >= rows) return;
    unsigned k = tid >> 4, c = tid & 15u;
    unsigned l = k / TOPK, slot = k % TOPK;
    unsigned idx = P.order[l * TOPK + slot];

    const float* a = P.anch[l] + (size_t)idx * 4u;
    const float* r = P.reg[l] + (size_t)idx * 8u;
    float x1 = a[0], y1 = a[1], x2 = a[2], y2 = a[3];
    float w = x2 - x1, h = y2 - y1;
    float cx = x1 + 0.5f * w, cy = y1 + 0.5f * h;
    float pcx = cx + r[0] * w;
    float pcy = cy + r[1] * h;
    float pw = w * expf(fminf(r[2], MAXD));
    float ph = h * expf(fminf(r[3], MAXD));
    float logit = P.cls[l][(size_t)idx * 16u + c];
    float score = 1.0f / (1.0f + expf(-logit));

    float* o = out + (size_t)tid * 6u;
    o[0] = pcx - 0.5f * pw;
    o[1] = pcy - 0.5f * ph;
    o[2] = pcx + 0.5f * pw;
    o[3] = pcy + 0.5f * ph;
    o[4] = score;
    o[5] = (float)(c + 1u);
}

extern "C" void kernel_launch(void* const* d_in, const int* in_sizes, int n_in,
                              void* d_out, int out_size, void* d_ws, size_t ws_size,
                              hipStream_t stream) {
    Params P;
    unsigned nmax = 0, off = 0;
    for (int l = 0; l < NLEV; ++l) {
        P.anch[l] = (const float*)d_in[3 * l + 0];
        P.cls[l]  = (const float*)d_in[3 * l + 1];
        P.reg[l]  = (const float*)d_in[3 * l + 2];
        unsigned n = (unsigned)in_sizes[3 * l + 0] / 4u;
        P.n[l] = n;
        P.monoOff[l] = off;
        off += n;
        if (n > nmax) nmax = n;
    }

    char* ws = (char*)d_ws;
    size_t o = 0;
    P.mono  = (unsigned*)(ws + o);            o += (size_t)off * 4;         o = (o + 255) & ~(size_t)255;
    P.hist  = (unsigned*)(ws + o);            o += NLEV * 4 * 256 * 4;      o = (o + 255) & ~(size_t)255;
    P.state = (unsigned*)(ws + o);            o += NLEV * 4 * 4;            o = (o + 255) & ~(size_t)255;
    P.selA  = (unsigned long long*)(ws + o);  o += NLEV * 1024 * 8;         o = (o + 255) & ~(size_t)255;
    P.selE  = (unsigned*)(ws + o);            o += NLEV * 1024 * 4;         o = (o + 255) & ~(size_t)255;
    P.order = (unsigned*)(ws + o);            o += NLEV * TOPK * 4;

    dim3 grid((nmax + 255u) / 256u, NLEV);

    k_init<<<1, 256, 0, stream>>>(P);
    k_ruler<<<grid, 256, 0, stream>>>(P);
    for (int p = 0; p < 4; ++p) {
        k_hist<<<grid, 256, 0, stream>>>(P, p);
        k_pick<<<NLEV, 1, 0, stream>>>(P, p);
    }
    k_compact<<<grid, 256, 0, stream>>>(P);
    k_sort<<<NLEV, 1024, 0, stream>>>(P);

    unsigned rows = (unsigned)(out_size / 6);
    k_out<<<(rows + 255u) / 256u, 256, 0, stream>>>(P, (float*)d_out, rows);
}